// GODEEncodingLayer_28243704939345
// MI455X (gfx1250) — compile-verified
//
#include <hip/hip_runtime.h>
#include <hip/hip_bf16.h>

// ---------------------------------------------------------------------------
// 2-layer GCN (PyG GCNConv semantics) for MI455X / gfx1250.
//   h  = tanh( Dinv (A+I) Dinv (x @ W1) + b1 )
//   out= sigmoid( Dinv (A+I) Dinv (h @ W2) + b2 )
// GEMMs run on the matrix pipe via V_WMMA_F32_16X16X4_F32 (exact fp32).
// Aggregation is wave-cooperative gather/atomic-scatter (L2-resident).
// ---------------------------------------------------------------------------

typedef float v2f __attribute__((ext_vector_type(2)));
typedef float v8f __attribute__((ext_vector_type(8)));

#define GCN_D 64  // both F_IN and D_OUT are 64

// ---- elementwise helpers ---------------------------------------------------

__global__ void k_zero_f32(float* __restrict__ p, int n) {
    int i = blockIdx.x * blockDim.x + threadIdx.x;
    if (i < n) p[i] = 0.0f;
}

// deg[col[e]] += ew[e]   (self loops handled analytically: deg_total = deg + 1)
__global__ void k_deg_scatter(const int* __restrict__ col,
                              const float* __restrict__ ew,
                              float* __restrict__ deg, int nedge) {
    int e = blockIdx.x * blockDim.x + threadIdx.x;
    if (e < nedge) atomicAdd(&deg[col[e]], ew[e]);
}

// dinv[i] = rsqrt(deg[i] + 1)   (+1 = self loop weight; always > 0)
__global__ void k_dinv(const float* __restrict__ deg,
                       float* __restrict__ dinv, int n) {
    int i = blockIdx.x * blockDim.x + threadIdx.x;
    if (i < n) dinv[i] = rsqrtf(deg[i] + 1.0f);
}

// agg[i,:] = xw[i,:] * dinv[i]^2   (self-loop term; also initializes agg)
__global__ void k_agg_init(const float* __restrict__ xw,
                           const float* __restrict__ dinv,
                           float* __restrict__ agg, int total) {
    int i = blockIdx.x * blockDim.x + threadIdx.x;
    if (i < total) {
        float d = dinv[i >> 6];  // node = i / 64
        agg[i] = xw[i] * d * d;
    }
}

// mode 0: out = tanh(agg + bias[f]);  mode 1: out = sigmoid(agg + bias[f])
__global__ void k_bias_act(const float* __restrict__ agg,
                           const float* __restrict__ bias,
                           float* __restrict__ out, int total, int mode) {
    int i = blockIdx.x * blockDim.x + threadIdx.x;
    if (i < total) {
        float v = agg[i] + bias[i & (GCN_D - 1)];
        out[i] = (mode == 0) ? tanhf(v) : (1.0f / (1.0f + __expf(-v)));
    }
}

// ---- edge scatter: one wave handles 32 edges cooperatively -----------------
// Lanes first load 32 edges' (row,col,norm) coalesced, then broadcast each
// edge via shuffles; every lane contributes 2 coalesced float atomics/edge.
__global__ void k_edge_scatter(const int* __restrict__ rowi,
                               const int* __restrict__ coli,
                               const float* __restrict__ ew,
                               const float* __restrict__ dinv,
                               const float* __restrict__ xw,
                               float* __restrict__ agg, int nedge) {
    const int lane  = threadIdx.x & 31;
    const int wave  = (blockIdx.x * (blockDim.x >> 5)) + (threadIdx.x >> 5);
    const int ebase = wave << 5;  // 32 edges per wave
    if (ebase >= nedge) return;

    int   r = 0, c = 0;
    float nrm = 0.0f;
    int   e = ebase + lane;
    if (e < nedge) {
        r = rowi[e];
        c = coli[e];
        nrm = dinv[r] * ew[e] * dinv[c];
    }
    const int cnt = min(32, nedge - ebase);
    for (int j = 0; j < cnt; ++j) {
        int   rj = __shfl(r, j);
        int   cj = __shfl(c, j);
        float nj = __shfl(nrm, j);
        const float* __restrict__ src = xw + (size_t)rj * GCN_D;
        float* __restrict__ dst = agg + (size_t)cj * GCN_D;
        atomicAdd(&dst[lane],        src[lane]        * nj);
        atomicAdd(&dst[lane + 32],   src[lane + 32]   * nj);
    }
}

// ---- fp32 WMMA GEMM: out[M x 64] = A[M x 64] @ W[64 x 64] ------------------
// One wave per 16x16 output tile; 8 waves/block -> 32 rows x 64 cols / block.
// K loop: 16 x V_WMMA_F32_16X16X4_F32.
// ISA fp32 fragment layouts (CDNA5 ISA §7.12.2):
//   A 16x4 : lane L<16  -> {A[L][k], A[L][k+1]},  lane L>=16 -> {A[L-16][k+2], A[L-16][k+3]}
//   B 4x16 : lane L<16  -> {B[k][L], B[k+1][L]},  lane L>=16 -> {B[k+2][L-16], B[k+3][L-16]}
//   C 16x16: VGPR r -> row (r + 8*hi), col = lane&15
__global__ void gcn_gemm64_wmma(const float* __restrict__ A,
                                const float* __restrict__ W,
                                float* __restrict__ out, int nrows) {
    const int wave = threadIdx.x >> 5;                 // 0..7
    const int lane = threadIdx.x & 31;
    const int row0 = blockIdx.x * 32 + (wave >> 2) * 16;
    const int col0 = (wave & 3) * 16;
    if (row0 >= nrows) return;
    const int lh = lane & 15;
    const int hi = lane >> 4;                          // 0 or 1

    v8f c = {};
#pragma unroll
    for (int k = 0; k < GCN_D; k += 4) {
        const int ka = k + 2 * hi;                     // this lane-half's K pair
        const float* __restrict__ ap = A + (size_t)(row0 + lh) * GCN_D + ka;
        v2f a;
        a.x = ap[0];
        a.y = ap[1];
        v2f b;
        b.x = W[(size_t)ka * GCN_D + col0 + lh];
        b.y = W[(size_t)(ka + 1) * GCN_D + col0 + lh];
        c = __builtin_amdgcn_wmma_f32_16x16x4_f32(
                /*neg_a=*/false, a, /*neg_b=*/false, b,
                /*c_mod=*/(short)0, c, /*reuse_a=*/false, /*reuse_b=*/false);
    }
#pragma unroll
    for (int r = 0; r < 8; ++r) {
        out[(size_t)(row0 + r + 8 * hi) * GCN_D + col0 + lh] = c[r];
    }
}

// ---------------------------------------------------------------------------

extern "C" void kernel_launch(void* const* d_in, const int* in_sizes, int n_in,
                              void* d_out, int out_size, void* d_ws, size_t ws_size,
                              hipStream_t stream) {
    const float* x  = (const float*)d_in[0];   // [N, 64]
    const int*   ei = (const int*)d_in[1];     // [2, E] (row-major: row then col)
    const float* ew = (const float*)d_in[2];   // [E]
    const float* W1 = (const float*)d_in[3];   // [64, 64]
    const float* b1 = (const float*)d_in[4];   // [64]
    const float* W2 = (const float*)d_in[5];   // [64, 64]
    const float* b2 = (const float*)d_in[6];   // [64]

    const int n = in_sizes[0] / GCN_D;         // 100000
    const int e = in_sizes[2];                 // 1250000
    const int* rowi = ei;                      // source
    const int* coli = ei + e;                  // target

    // workspace layout
    float* deg  = (float*)d_ws;                // [N]
    float* dinv = deg + n;                     // [N]
    float* buf0 = dinv + n;                    // [N*64]  xw / hw2
    float* buf1 = buf0 + (size_t)n * GCN_D;    // [N*64]  agg / h (ping-pong)

    const int total = n * GCN_D;
    const int B256n  = (n + 255) / 256;
    const int B256e  = (e + 255) / 256;
    const int B256t  = (total + 255) / 256;
    const int gemmB  = (n + 31) / 32;
    const int scatB  = ((e + 31) / 32 + 7) / 8;  // 8 waves/block, 32 edges/wave

    // normalization: deg -> dinv
    k_zero_f32   <<<B256n, 256, 0, stream>>>(deg, n);
    k_deg_scatter<<<B256e, 256, 0, stream>>>(coli, ew, deg, e);
    k_dinv       <<<B256n, 256, 0, stream>>>(deg, dinv, n);

    // layer 1
    gcn_gemm64_wmma<<<gemmB, 256, 0, stream>>>(x, W1, buf0, n);          // xw
    k_agg_init     <<<B256t, 256, 0, stream>>>(buf0, dinv, buf1, total); // self loops
    k_edge_scatter <<<scatB, 256, 0, stream>>>(rowi, coli, ew, dinv, buf0, buf1, e);
    k_bias_act     <<<B256t, 256, 0, stream>>>(buf1, b1, buf1, total, 0); // h = tanh(.)

    // layer 2
    gcn_gemm64_wmma<<<gemmB, 256, 0, stream>>>(buf1, W2, buf0, n);       // hw2
    k_agg_init     <<<B256t, 256, 0, stream>>>(buf0, dinv, buf1, total);
    k_edge_scatter <<<scatB, 256, 0, stream>>>(rowi, coli, ew, dinv, buf0, buf1, e);
    k_bias_act     <<<B256t, 256, 0, stream>>>(buf1, b2, (float*)d_out, total, 1);
}